// DetBenchPredict_2783138808141
// MI455X (gfx1250) — compile-verified
//
#include <hip/hip_runtime.h>
#include <math.h>

typedef unsigned int u32;
typedef unsigned long long u64;

#define NCLS   90
#define MAXK   5000
#define MAXDET 100
#define BATCHN 4
#define NSCORE 4419360   // 810 * (4096+1024+256+64+16)
#define CLS_C3 4860      // 3 * 9 * 90 * 2
#define BOX_C3 216       // 3 * 9 * 4  * 2

#ifndef __has_builtin
#define __has_builtin(x) 0
#endif

#if __has_builtin(__builtin_amdgcn_tensor_load_to_lds) && __has_builtin(__builtin_amdgcn_s_wait_tensorcnt)
#define USE_TDM 1
#else
#define USE_TDM 0
#endif

#if __has_builtin(__builtin_amdgcn_wmma_f32_16x16x4_f32)
#define USE_WMMA_SCAN 1
#else
#define USE_WMMA_SCAN 0
#endif

typedef unsigned int uint32x4 __attribute__((ext_vector_type(4)));
typedef int          int32x8  __attribute__((ext_vector_type(8)));
typedef int          int32x4  __attribute__((ext_vector_type(4)));
typedef float        v2f      __attribute__((ext_vector_type(2)));
typedef float        v8f      __attribute__((ext_vector_type(8)));

// Order-preserving float -> uint key (ascending float => ascending key).
__device__ __forceinline__ u32 fkey(float f) {
  u32 u = __float_as_uint(f);
  return (u & 0x80000000u) ? ~u : (u | 0x80000000u);
}

__device__ __forceinline__ float sigmoidf_(float x) { return 1.0f / (1.0f + expf(-x)); }

#if USE_TDM
// 2-D tile load via Tensor Data Mover: rows = channels (stride = H*W elems),
// row length = tile_w f32 elements, packed contiguously into LDS.
// D# field packing per CDNA5 ISA.
__device__ __forceinline__ void tdm_load_2d(u64 gaddr, u32 lds_byte_off,
                                            int tile_w, int rows, int row_stride_elems) {
  uint32x4 g0;
  g0[0] = 1u;                                               // count=1, gather off
  g0[1] = lds_byte_off;                                     // lds_addr
  g0[2] = (u32)(gaddr & 0xffffffffu);                       // global_addr[31:0]
  g0[3] = (u32)((gaddr >> 32) & 0x1ffffffu) | (2u << 30);   // global_addr[56:32] | type=2
  int32x8 g1;
  g1[0] = 0x20000;                                          // data_size=2 (4 bytes)
  g1[1] = (int)(((u32)tile_w & 0xffffu) << 16);             // tensor_dim0 lo16
  g1[2] = (int)((((u32)rows & 0xffffu) << 16));             // tensor_dim1 lo16 (dim0 hi=0)
  g1[3] = (int)(((u32)tile_w & 0xffffu) << 16);             // tile_dim0 (dim1 hi=0)
  g1[4] = rows & 0xffff;                                    // tile_dim1 (tile_dim2=0)
  g1[5] = row_stride_elems;                                 // tensor_dim0_stride[31:0]
  g1[6] = 0;
  g1[7] = 0;
  int32x4 z4 = {0, 0, 0, 0};
#if __clang_major__ >= 23
  int32x8 z8 = {0, 0, 0, 0, 0, 0, 0, 0};
  __builtin_amdgcn_tensor_load_to_lds(g0, g1, z4, z4, z8, 0);
#else
  __builtin_amdgcn_tensor_load_to_lds(g0, g1, z4, z4, 0);
#endif
}
#endif

// ---------------- kZ: zero hist/subhist/meta/counters ----------------
__global__ void kZ_zero(u32* __restrict__ p, int n) {
  int i = blockIdx.x * 256 + threadIdx.x;
  if (i < n) p[i] = 0;
}

// ---------------- kA: GMM weighted-mean scores + level-1 histogram ----------------
__global__ void __launch_bounds__(256) kA_cls_scores(
    const float* __restrict__ c0p, const float* __restrict__ c1p,
    const float* __restrict__ c2p, const float* __restrict__ c3p,
    const float* __restrict__ c4p,
    float* __restrict__ scores, u32* __restrict__ ghist) {
  __shared__ float tile[2 * 12 * 256];   // [mean rows | weight rows], 24 KB
  __shared__ u32 hsh[1024];
  const int fs2tab[5]  = {4096, 1024, 256, 64, 16};
  const int lofftab[5] = {0, 3317760, 4147200, 4354560, 4406400};
  int z = blockIdx.z;
  int b = z / 5, l = z - b * 5;
  int fs2 = fs2tab[l];
  int p0 = (int)(blockIdx.y << 8);
  if (p0 >= fs2) return;                 // uniform per block
  int P = fs2 - p0; if (P > 256) P = 256;
  int c0 = (int)blockIdx.x * 6;          // 135 chunks * 6 = 810 exactly
  int tid = threadIdx.x;
  const float* src = (l == 0) ? c0p : (l == 1) ? c1p : (l == 2) ? c2p : (l == 3) ? c3p : c4p;

  for (int k = tid; k < 1024; k += 256) hsh[k] = 0;

#if USE_TDM
  if (tid < 32) {  // wave 0 only issues the TDM transfers
    u32 ldsb = (u32)(uintptr_t)(void*)tile;  // low 32 bits of generic shared addr = LDS offset
    u64 base = (u64)(uintptr_t)src;
    u64 ga0 = base + 4ull * (((u64)b * CLS_C3 + 2u * (u32)c0) * (u64)fs2 + (u64)p0);           // means
    u64 ga1 = base + 4ull * (((u64)b * CLS_C3 + 3240u + 2u * (u32)c0) * (u64)fs2 + (u64)p0);   // weights
    tdm_load_2d(ga0, ldsb, P, 12, fs2);
    tdm_load_2d(ga1, ldsb + 12u * 256u * 4u, P, 12, fs2);
    __builtin_amdgcn_s_wait_tensorcnt(0);
  }
#else
  for (int t = 0; t < 2; ++t) {
    int cb = (t ? 3240 : 0) + 2 * c0;
    for (int idx = tid; idx < 12 * P; idx += 256) {
      int r = idx / P, x = idx - r * P;
      tile[t * 3072 + r * P + x] = src[((size_t)b * CLS_C3 + cb + r) * (size_t)fs2 + p0 + x];
    }
  }
#endif
  __syncthreads();

  if (tid < P) {
    size_t obase = (size_t)b * NSCORE + (size_t)lofftab[l] + (size_t)p0 + (size_t)tid;
    for (int ci = 0; ci < 6; ++ci) {
      float m0 = tile[(2 * ci) * P + tid];
      float m1 = tile[(2 * ci + 1) * P + tid];
      float w0 = tile[3072 + (2 * ci) * P + tid];
      float w1 = tile[3072 + (2 * ci + 1) * P + tid];
      float mx = fmaxf(w0, w1);
      float e0 = expf(w0 - mx), e1 = expf(w1 - mx);
      float wm = (e0 * m0 + e1 * m1) / (e0 + e1);
      scores[obase + (size_t)(c0 + ci) * (size_t)fs2] = wm;
      atomicAdd(&hsh[fkey(wm) >> 22], 1u);
    }
  }
  __syncthreads();
  for (int k = tid; k < 1024; k += 256) {
    u32 v = hsh[k];
    if (v) atomicAdd(&ghist[b * 1024 + k], v);
  }
}

// ---------------- kB: threshold via WMMA suffix-scan of 1024-bin histogram ----------
// Suffix sum S = L * X with L upper-triangular ones, computed on the matrix pipe:
// level-1: 4 batches x (16 cols x 16 rows), K sliced by 4 -> 16 v_wmma_f32_16x16x4_f32.
// level-2: scan of the 64 segment totals (4 cols) -> 4 more WMMAs, then 4-way combine.
// level==0: derive bucket T for top-MAXK; level==1: refine inside bucket T.
__global__ void __launch_bounds__(32) kB_thresh(const u32* __restrict__ histb,
                                                u32* __restrict__ meta, int level) {
  int b = blockIdx.x;
  const u32* h = histb + b * 1024;
  u32 need = (level == 0) ? (u32)MAXK : meta[b * 8 + 5];
  int T; u32 above;
#if USE_WMMA_SCAN
  __shared__ float S[1024];    // inclusive within-segment suffix sums, then final S
  __shared__ float tot[64];    // segment totals
  __shared__ float stot[64];   // within-group inclusive suffix of totals
  int lane = threadIdx.x;
  int hl = lane >> 4;          // 0: lanes 0-15, 1: lanes 16-31
  int lm = lane & 15;

  // ---- level-1: per-16-segment inclusive suffix scans (elements k = n*16 + m) ----
  for (int g = 0; g < 4; ++g) {           // columns n = 16g .. 16g+15
    v8f acc = {};
#pragma unroll
    for (int bk = 0; bk < 4; ++bk) {
      int k0 = 4 * bk + (hl ? 2 : 0);
      v2f a;                               // A[m][k] = (k >= m), m = lm
      a.x = (k0 >= lm) ? 1.0f : 0.0f;
      a.y = (k0 + 1 >= lm) ? 1.0f : 0.0f;
      v2f bv;                              // B[k][n] = h[(16g+n)*16 + k], n = lm
      bv.x = (float)h[(16 * g + lm) * 16 + k0];
      bv.y = (float)h[(16 * g + lm) * 16 + k0 + 1];
      acc = __builtin_amdgcn_wmma_f32_16x16x4_f32(false, a, false, bv,
                                                  (short)0, acc, false, false);
    }
#pragma unroll
    for (int j = 0; j < 8; ++j) {          // D: vgpr j = row m (lanes<16) / m+8 (lanes>=16)
      int m = j + (hl ? 8 : 0);
      S[(16 * g + lm) * 16 + m] = acc[j];
    }
    if (hl == 0) tot[16 * g + lm] = acc[0];  // Y[0][n] = segment total
  }
  __syncthreads();

  // ---- level-2: suffix scan of 64 totals, grouped as 4 columns of 16 ----
  {
    v8f acc = {};
#pragma unroll
    for (int bk = 0; bk < 4; ++bk) {
      int k0 = 4 * bk + (hl ? 2 : 0);
      v2f a;
      a.x = (k0 >= lm) ? 1.0f : 0.0f;
      a.y = (k0 + 1 >= lm) ? 1.0f : 0.0f;
      v2f bv;
      bv.x = (lm < 4) ? tot[lm * 16 + k0] : 0.0f;
      bv.y = (lm < 4) ? tot[lm * 16 + k0 + 1] : 0.0f;
      acc = __builtin_amdgcn_wmma_f32_16x16x4_f32(false, a, false, bv,
                                                  (short)0, acc, false, false);
    }
#pragma unroll
    for (int j = 0; j < 8; ++j) {
      int m = j + (hl ? 8 : 0);
      if (lm < 4) stot[lm * 16 + m] = acc[j];
    }
  }
  __syncthreads();

  // ---- combine: S[k] += (suffix of later segments) ----
  float sup1 = stot[16], sup2 = stot[32], sup3 = stot[48];
  float osup[4] = {sup1 + sup2 + sup3, sup2 + sup3, sup3, 0.0f};
  for (int k = lane; k < 1024; k += 32) {
    int n = k >> 4, m2 = n & 15, col = n >> 4;
    S[k] += (stot[col * 16 + m2] - tot[n]) + osup[col];
  }
  __syncthreads();

  // ---- find largest T with S[T] >= need ----
  float needf = (float)need;
  int bestT = -1;
  for (int k = lane; k < 1024; k += 32)
    if (S[k] >= needf && k > bestT) bestT = k;
  for (int m = 16; m > 0; m >>= 1) {
    int o = __shfl_xor(bestT, m, 32);
    if (o > bestT) bestT = o;
  }
  if (lane != 0) return;
  T = (bestT < 0) ? 0 : bestT;
  above = (u32)(S[T]) - h[T];   // strictly-above-T count
#else
  if (threadIdx.x != 0) return;
  u32 cum = 0; T = 0; above = 0; bool found = false;
  for (int k = 1023; k >= 0 && !found; --k) {
    u32 hk = h[k];
    if (cum + hk >= need) { T = k; above = cum; found = true; }
    else cum += hk;
  }
  if (!found) { T = 0; above = cum; }
#endif
  if (level == 0) {
    meta[b * 8 + 0] = (u32)T;
    meta[b * 8 + 4] = above;
    meta[b * 8 + 5] = (u32)MAXK - above;
  } else {
    u32 above1 = meta[b * 8 + 4];
    meta[b * 8 + 1] = (u32)T;
    meta[b * 8 + 2] = above1 + above;   // definitely-in count
    meta[b * 8 + 3] = need - above;     // ration from (T, T2) bucket
  }
}

// ---------------- kS: sub-histogram (key bits 21:12) within bucket T ----------------
__global__ void kS_subhist(const float* __restrict__ scores, const u32* __restrict__ meta,
                           u32* __restrict__ subhist) {
  size_t gid = (size_t)blockIdx.x * 256 + threadIdx.x;
  if (gid >= (size_t)BATCHN * NSCORE) return;
  int b = (int)(gid / NSCORE);
  u32 key = fkey(scores[gid]);
  if ((key >> 22) != meta[b * 8 + 0]) return;
  atomicAdd(&subhist[b * 1024 + ((key >> 12) & 0x3ffu)], 1u);
}

// ---------------- kB2: compact top-5000 candidates ----------------
__global__ void kB2_select(const float* __restrict__ scores, const u32* __restrict__ meta,
                           u32* __restrict__ cnts, u32* __restrict__ cidx,
                           float* __restrict__ craw) {
  size_t gid = (size_t)blockIdx.x * 256 + threadIdx.x;
  if (gid >= (size_t)BATCHN * NSCORE) return;
  int b = (int)(gid / NSCORE);
  u32 i = (u32)(gid - (size_t)b * NSCORE);
  float s = scores[gid];
  u32 key = fkey(s);
  u32 k10 = key >> 22;
  u32 T = meta[b * 8 + 0];
  if (k10 < T) return;
  u32 slot;
  if (k10 > T) {
    slot = atomicAdd(&cnts[b * 2 + 0], 1u);
  } else {
    u32 k12 = (key >> 12) & 0x3ffu;
    u32 T2 = meta[b * 8 + 1];
    if (k12 < T2) return;
    if (k12 > T2) {
      slot = atomicAdd(&cnts[b * 2 + 0], 1u);
    } else {
      u32 q = atomicAdd(&cnts[b * 2 + 1], 1u);
      if (q >= meta[b * 8 + 3]) return;
      slot = meta[b * 8 + 2] + q;
    }
  }
  cidx[(size_t)b * MAXK + slot] = i;
  craw[(size_t)b * MAXK + slot] = s;
}

// ---------------- kC: per-candidate extras + box decode ----------------
__global__ void kC_decode(
    const float* __restrict__ cls0, const float* __restrict__ cls1,
    const float* __restrict__ cls2, const float* __restrict__ cls3,
    const float* __restrict__ cls4,
    const float* __restrict__ box0, const float* __restrict__ box1,
    const float* __restrict__ box2, const float* __restrict__ box3,
    const float* __restrict__ box4,
    const float* __restrict__ anchors, const float* __restrict__ img_scale,
    const float* __restrict__ img_size,
    const u32* __restrict__ cidx, const float* __restrict__ craw,
    float4* __restrict__ cboxes, float* __restrict__ cscore,
    int* __restrict__ cclass, float4* __restrict__ cextras) {
  int gid = blockIdx.x * 256 + threadIdx.x;
  if (gid >= BATCHN * MAXK) return;
  int b = gid / MAXK;
  u32 i = cidx[gid];
  float raw = craw[gid];
  const u32 cum[5]  = {3317760u, 4147200u, 4354560u, 4406400u, 4419360u};
  const int sht[5]  = {12, 10, 8, 6, 4};
  const u32 aoff[5] = {0u, 36864u, 46080u, 48384u, 48960u};
  int l = 0;
  while (l < 4 && i >= cum[l]) ++l;
  u32 rem = i - (l ? cum[l - 1] : 0u);
  int sh = sht[l]; int fs2 = 1 << sh;
  u32 c = rem >> sh; u32 p = rem & (u32)(fs2 - 1);
  u32 a = c / 90u; u32 cls = c - a * 90u;
  u32 n = aoff[l] + p * 9u + a;
  const float* csrc = (l == 0) ? cls0 : (l == 1) ? cls1 : (l == 2) ? cls2 : (l == 3) ? cls3 : cls4;
  const float* bsrc = (l == 0) ? box0 : (l == 1) ? box1 : (l == 2) ? box2 : (l == 3) ? box3 : box4;

  // cls ua / ue at (n, cls)
  size_t cb = ((size_t)b * CLS_C3 + 2u * c) * (size_t)fs2 + p;
  float m0 = csrc[cb], m1 = csrc[cb + fs2];
  float v0 = csrc[cb + (size_t)1620 * fs2], v1 = csrc[cb + (size_t)1620 * fs2 + fs2];
  float w0 = csrc[cb + (size_t)3240 * fs2], w1 = csrc[cb + (size_t)3240 * fs2 + fs2];
  float mx = fmaxf(w0, w1);
  float e0 = expf(w0 - mx), e1 = expf(w1 - mx);
  float inv = 1.0f / (e0 + e1);
  float ww0 = e0 * inv, ww1 = e1 * inv;
  float wm = ww0 * m0 + ww1 * m1;
  float cls_ua = ww0 * sigmoidf_(v0) + ww1 * sigmoidf_(v1);
  float d0 = m0 - wm, d1 = m1 - wm;
  float cls_ue = ww0 * d0 * d0 + ww1 * d1 * d1;

  // box wm/ua/ue over 4 coords
  float bt[4]; float bua = -1e30f, bue = -1e30f;
  for (int j = 0; j < 4; ++j) {
    size_t bb = ((size_t)b * BOX_C3 + 2u * (a * 4u + j)) * (size_t)fs2 + p;
    float bm0 = bsrc[bb], bm1 = bsrc[bb + fs2];
    float bv0 = bsrc[bb + (size_t)72 * fs2], bv1 = bsrc[bb + (size_t)72 * fs2 + fs2];
    float bw0 = bsrc[bb + (size_t)144 * fs2], bw1 = bsrc[bb + (size_t)144 * fs2 + fs2];
    float bmx = fmaxf(bw0, bw1);
    float be0 = expf(bw0 - bmx), be1 = expf(bw1 - bmx);
    float binv = 1.0f / (be0 + be1);
    float q0 = be0 * binv, q1 = be1 * binv;
    float bwm = q0 * bm0 + q1 * bm1;
    bt[j] = bwm;
    float ua = q0 * sigmoidf_(bv0) + q1 * sigmoidf_(bv1);
    float f0 = bm0 - bwm, f1 = bm1 - bwm;
    float ue = q0 * f0 * f0 + q1 * f1 * f1;
    bua = fmaxf(bua, ua); bue = fmaxf(bue, ue);
  }

  float a0 = anchors[n * 4 + 0], a1 = anchors[n * 4 + 1];
  float a2 = anchors[n * 4 + 2], a3 = anchors[n * 4 + 3];
  float ya = 0.5f * (a0 + a2), xa = 0.5f * (a1 + a3);
  float ha = a2 - a0, wa = a3 - a1;
  float yc = bt[0] * ha + ya, xc = bt[1] * wa + xa;
  float hh = expf(bt[2]) * ha, wd = expf(bt[3]) * wa;
  float sc = img_scale[b];
  float H = img_size[b * 2 + 0], W = img_size[b * 2 + 1];
  float hiX = W * sc, hiY = H * sc;
  float4 bxo;
  bxo.x = fminf(fmaxf((xc - 0.5f * wd) * sc, 0.0f), hiX);
  bxo.y = fminf(fmaxf((yc - 0.5f * hh) * sc, 0.0f), hiY);
  bxo.z = fminf(fmaxf((xc + 0.5f * wd) * sc, 0.0f), hiX);
  bxo.w = fminf(fmaxf((yc + 0.5f * hh) * sc, 0.0f), hiY);
  cboxes[gid] = bxo;
  cscore[gid] = sigmoidf_(raw);
  cclass[gid] = (int)cls;
  float4 exo; exo.x = cls_ua; exo.y = cls_ue; exo.z = bua; exo.w = bue;
  cextras[gid] = exo;
}

// ---------------- kD: greedy class-aware NMS (one block per image) ----------------
__global__ void __launch_bounds__(256) kD_nms(
    const float4* __restrict__ cboxes, const float* __restrict__ cscore,
    const int* __restrict__ cclass, const float4* __restrict__ cextras,
    float* __restrict__ out) {
  __shared__ float act[MAXK];
  __shared__ float rv[8];
  __shared__ int ri[8];
  __shared__ float bb[4];
  __shared__ int bcj;
  int b = blockIdx.x;
  int tid = threadIdx.x;
  const float4* bx = cboxes + (size_t)b * MAXK;
  const int* cl = cclass + (size_t)b * MAXK;
  for (int i = tid; i < MAXK; i += 256) act[i] = cscore[(size_t)b * MAXK + i];
  __syncthreads();
  for (int it = 0; it < MAXDET; ++it) {
    // strided per-thread argmax over active scores
    float bvv = -1e30f; int bii = 0;
    for (int i = tid; i < MAXK; i += 256) {
      float v = act[i];
      if (v > bvv) { bvv = v; bii = i; }
    }
    // wave32 butterfly reduction (no barriers)
    for (int m = 16; m > 0; m >>= 1) {
      float ov = __shfl_xor(bvv, m, 32);
      int oi = __shfl_xor(bii, m, 32);
      if (ov > bvv || (ov == bvv && oi < bii)) { bvv = ov; bii = oi; }
    }
    if ((tid & 31) == 0) { rv[tid >> 5] = bvv; ri[tid >> 5] = bii; }
    __syncthreads();
    if (tid == 0) {
      float fv = rv[0]; int fj = ri[0];
      for (int wv = 1; wv < 8; ++wv) {
        float v2 = rv[wv]; int i2 = ri[wv];
        if (v2 > fv || (v2 == fv && i2 < fj)) { fv = v2; fj = i2; }
      }
      float4 bj = bx[fj];
      int cj = cl[fj];
      float4 ex = cextras[(size_t)b * MAXK + fj];
      float* o = out + ((size_t)b * MAXDET + it) * 10;
      o[0] = bj.x; o[1] = bj.y; o[2] = bj.z; o[3] = bj.w;
      o[4] = fv;   o[5] = (float)cj;
      o[6] = ex.x; o[7] = ex.y; o[8] = ex.z; o[9] = ex.w;
      bb[0] = bj.x; bb[1] = bj.y; bb[2] = bj.z; bb[3] = bj.w; bcj = cj;
      act[fj] = -1.0f;
    }
    __syncthreads();
    float b0 = bb[0], b1 = bb[1], b2 = bb[2], b3 = bb[3];
    int cj = bcj;
    float areaB = (b2 - b0) * (b3 - b1);
    for (int i = tid; i < MAXK; i += 256) {
      if (cl[i] != cj) continue;
      float4 ci = bx[i];
      float areaI = (ci.z - ci.x) * (ci.w - ci.y);
      float ltx = fmaxf(b0, ci.x), lty = fmaxf(b1, ci.y);
      float rbx = fminf(b2, ci.z), rby = fminf(b3, ci.w);
      float iw = fmaxf(rbx - ltx, 0.0f), ih = fmaxf(rby - lty, 0.0f);
      float inter = iw * ih;
      float iou = inter / (areaB + areaI - inter + 1e-8f);
      if (iou > 0.5f) act[i] = -1.0f;
    }
    __syncthreads();
  }
}

// ---------------- host-side launcher ----------------
extern "C" void kernel_launch(void* const* d_in, const int* in_sizes, int n_in,
                              void* d_out, int out_size, void* d_ws, size_t ws_size,
                              hipStream_t stream) {
  (void)in_sizes; (void)n_in; (void)out_size; (void)ws_size;
  // dict order: cls0, box0, cls1, box1, ..., cls4, box4, anchors, img_scale, img_size
  const float* cls[5] = {(const float*)d_in[0], (const float*)d_in[2], (const float*)d_in[4],
                         (const float*)d_in[6], (const float*)d_in[8]};
  const float* box[5] = {(const float*)d_in[1], (const float*)d_in[3], (const float*)d_in[5],
                         (const float*)d_in[7], (const float*)d_in[9]};
  const float* anchors   = (const float*)d_in[10];
  const float* img_scale = (const float*)d_in[11];
  const float* img_size  = (const float*)d_in[12];

  char* ws = (char*)d_ws;
  float*  scores  = (float*)(ws + 0);            // 70,709,760 B
  u32*    hist    = (u32*)(ws + 70709760);       // 4*1024
  u32*    subhist = hist + 4096;                 // 4*1024
  u32*    meta    = subhist + 4096;              // 4*8
  u32*    cnts    = meta + 32;                   // 4*2
  u32*    cidx    = (u32*)(ws + 70742784);       // 4*5000
  float*  craw    = (float*)(ws + 70822784);
  float4* cboxes  = (float4*)(ws + 70902784);    // 4*5000 float4
  float*  cscore  = (float*)(ws + 71222784);
  int*    cclass  = (int*)(ws + 71302784);
  float4* cextras = (float4*)(ws + 71382784);    // 4*5000 float4

  kZ_zero<<<33, 256, 0, stream>>>(hist, 8232);   // hist + subhist + meta + counters

  dim3 gA(135, 16, 20);  // c-chunks x p-chunks x (b*5+level)
  kA_cls_scores<<<gA, 256, 0, stream>>>(cls[0], cls[1], cls[2], cls[3], cls[4], scores, hist);

  kB_thresh<<<4, 32, 0, stream>>>(hist, meta, 0);

  size_t tot = (size_t)BATCHN * NSCORE;
  unsigned nblk = (unsigned)((tot + 255) / 256);
  kS_subhist<<<nblk, 256, 0, stream>>>(scores, meta, subhist);
  kB_thresh<<<4, 32, 0, stream>>>(subhist, meta, 1);
  kB2_select<<<nblk, 256, 0, stream>>>(scores, meta, cnts, cidx, craw);

  kC_decode<<<(BATCHN * MAXK + 255) / 256, 256, 0, stream>>>(
      cls[0], cls[1], cls[2], cls[3], cls[4],
      box[0], box[1], box[2], box[3], box[4],
      anchors, img_scale, img_size, cidx, craw, cboxes, cscore, cclass, cextras);

  kD_nms<<<4, 256, 0, stream>>>(cboxes, cscore, cclass, cextras, (float*)d_out);
}